// RealAgnosticLongRangeResidualInteractionBlock_72816875537125
// MI455X (gfx1250) — compile-verified
//
#include <hip/hip_runtime.h>
#include <hip/hip_bf16.h>

typedef float v2f __attribute__((ext_vector_type(2)));
typedef float v4f __attribute__((ext_vector_type(4)));
typedef float v8f __attribute__((ext_vector_type(8)));

__device__ __forceinline__ v8f wmma4(v2f a, v2f b, v8f c) {
    // V_WMMA_F32_16X16X4_F32 : D(16x16 f32) = A(16x4) * B(4x16) + C
    return __builtin_amdgcn_wmma_f32_16x16x4_f32(
        false, a, false, b, (short)0, c, false, false);
}

__device__ __forceinline__ float silu_f(float x) {
    return x / (1.0f + __expf(-x));
}

// 16x16 output tile, K=64, A fragments from LDS (hb = &h[ll][2*half], row
// already selected), B from global with a depth-4 rotating register prefetch
// so each WMMA waits on loads issued 4 iterations earlier, not 0.
template<int LDB>
__device__ __forceinline__ v8f tile16(const float* hb, const float* __restrict__ Wb) {
    v8f acc = {};
    v2f b[4];
#pragma unroll
    for (int s = 0; s < 4; ++s) {
        b[s].x = Wb[(4 * s) * LDB];
        b[s].y = Wb[(4 * s) * LDB + LDB];
    }
#pragma unroll
    for (int i = 0; i < 16; ++i) {
        v2f av;
        av.x = hb[4 * i];
        av.y = hb[4 * i + 1];
        acc = wmma4(av, b[i & 3], acc);
        if (i + 4 < 16) {
            b[i & 3].x = Wb[(4 * (i + 4)) * LDB];
            b[i & 3].y = Wb[(4 * (i + 4)) * LDB + LDB];
        }
    }
    return acc;
}

// ---------------------------------------------------------------------------
// zero scratch accumulators
// ---------------------------------------------------------------------------
__global__ void zero_kernel(float* __restrict__ p, long n) {
    long i = (long)blockIdx.x * blockDim.x + threadIdx.x;
    long stride = (long)gridDim.x * blockDim.x;
    for (; i < n; i += stride) p[i] = 0.0f;
}

// ---------------------------------------------------------------------------
// LDS-staged f32 WMMA GEMM, N = 128 (8 column tiles = 8 waves / block),
// grid.x = M/16. B row stride fixed at 128 (row-major weights).
// A tile (16xK) staged once; B staged in 64x128 chunks via coalesced b128.
// Compute phase reads LDS only. D[m,n] = alpha * sum_k A*B ( / (dens[m]+1) )
// ---------------------------------------------------------------------------
template<int K, int ACS, int DCS, bool DENS>
__global__ __launch_bounds__(256) void gemm16_kernel(
    const float* __restrict__ A, long a_rs,
    const float* __restrict__ B,
    float* __restrict__ D, long d_rs,
    float alpha, const float* __restrict__ dens)
{
    __shared__ float sA[16][K + 4];     // +4 pad: conflict-free A-frag b64 loads
    __shared__ float sB[64][132];       // +4 pad

    const int tid = threadIdx.x;
    const int tm  = blockIdx.x;

    // ---- stage A (16 x K) ----
    if (ACS == 1) {
#pragma unroll
        for (int idx = tid; idx < 16 * K / 4; idx += 256) {
            const int row = idx / (K / 4);
            const int c4  = idx % (K / 4);
            ((v4f*)&sA[row][0])[c4] =
                *(const v4f*)(A + (long)(tm * 16 + row) * a_rs + c4 * 4);
        }
    } else {
        for (int idx = tid; idx < 16 * K; idx += 256) {
            const int row = idx / K;
            const int c   = idx % K;
            sA[row][c] = A[(long)(tm * 16 + row) * a_rs + (long)c * ACS];
        }
    }

    const int lane = tid & 31;
    const int half = lane >> 4;
    const int ll   = lane & 15;
    const int tn   = tid >> 5;
    const int col  = tn * 16 + ll;
    const float* hb = &sA[ll][2 * half];

    v8f acc = {};
#pragma unroll
    for (int ch = 0; ch < K / 64; ++ch) {
        __syncthreads();                 // sA ready / sB consumed
        {   // stage B rows [ch*64, ch*64+64)
            const int j  = tid >> 2;
            const int qd = tid & 3;
            const float* src = B + (long)(ch * 64 + j) * 128 + qd * 32;
            v4f* dst = (v4f*)&sB[j][qd * 32];
#pragma unroll
            for (int q = 0; q < 8; ++q) dst[q] = *(const v4f*)(src + q * 4);
        }
        __syncthreads();
#pragma unroll
        for (int i = 0; i < 16; ++i) {
            v2f av, bv;
            av.x = hb[ch * 64 + 4 * i];
            av.y = hb[ch * 64 + 4 * i + 1];
            bv.x = sB[4 * i + 2 * half][col];
            bv.y = sB[4 * i + 2 * half + 1][col];
            acc = wmma4(av, bv, acc);
        }
    }

#pragma unroll
    for (int r = 0; r < 8; ++r) {
        const int row = tm * 16 + half * 8 + r;
        float s = alpha;
        if (DENS) s /= (dens[row] + 1.0f);
        D[(long)row * d_rs + (long)col * DCS] = acc[r] * s;
    }
}

// ---------------------------------------------------------------------------
// Skip connection: sc0[n,w] = sum_{u,v} s[n,u] attrs[n,v] Wsk0[u,v,w] * norm
//                  sc1[n,w,m] likewise with v-features and Wsk1.
// K = 1280 reduction in (v outer, u inner) order; B rows staged into LDS in
// that permuted order (srcrow = ((c&1)*64+j)*10 + (c>>1)); per-lane A operands
// cached in registers. Compute phase: 2 v_mul + 2 ds_load + WMMA.
// grid = (625, 4): y==0 scalar part, y==1..3 vector component m = y-1.
// ---------------------------------------------------------------------------
__global__ __launch_bounds__(256) void skip_kernel(
    const float* __restrict__ node_feats, const float* __restrict__ node_attrs,
    const float* __restrict__ Wsk0, const float* __restrict__ Wsk1,
    float* __restrict__ out_sc)
{
    __shared__ float s_s[16][128];
    __shared__ float s_a[16][10];
    __shared__ float sB[64][132];

    const int comp = blockIdx.y;
    const int tm   = blockIdx.x;

    for (int t = threadIdx.x; t < 16 * 128; t += 256) {
        const int e = t >> 7, c = t & 127;
        const long n = (long)tm * 16 + e;
        s_s[e][c] = (comp == 0) ? node_feats[n * 512 + c]
                                : node_feats[n * 512 + 128 + c * 3 + (comp - 1)];
    }
    for (int t = threadIdx.x; t < 160; t += 256) {
        const int e = t / 10, c = t % 10;
        s_a[e][c] = node_attrs[((long)tm * 16 + e) * 10 + c];
    }
    __syncthreads();

    const int lane = threadIdx.x & 31;
    const int half = lane >> 4;
    const int ll   = lane & 15;
    const int tn   = threadIdx.x >> 5;   // 8 waves -> 8 column tiles (C=128)
    const int col  = tn * 16 + ll;
    const float* W = (comp == 0) ? Wsk0 : Wsk1;

    // Per-lane A operands: u values { 4w + 2*half, 4w + 2*half + 1 }, w=0..31
    float sr[64];
#pragma unroll
    for (int w = 0; w < 32; ++w) {
        sr[2 * w]     = s_s[ll][4 * w + 2 * half];
        sr[2 * w + 1] = s_s[ll][4 * w + 2 * half + 1];
    }
    float ar[10];
#pragma unroll
    for (int v = 0; v < 10; ++v) ar[v] = s_a[ll][v];

    v8f acc = {};
#pragma unroll
    for (int c = 0; c < 20; ++c) {
        __syncthreads();
        {   // stage permuted B rows: compute-order q = 64c + j
            // u = (q & 127), v = (q >> 7)  ->  srcrow = u*10 + v
            const int j  = threadIdx.x >> 2;
            const int qd = threadIdx.x & 3;
            const int srcrow = ((c & 1) * 64 + j) * 10 + (c >> 1);
            const float* src = W + (long)srcrow * 128 + qd * 32;
            v4f* dst = (v4f*)&sB[j][qd * 32];
#pragma unroll
            for (int q = 0; q < 8; ++q) dst[q] = *(const v4f*)(src + q * 4);
        }
        __syncthreads();
        const float aval = ar[c >> 1];
        const int   w0   = (c & 1) * 16;
#pragma unroll
        for (int i = 0; i < 16; ++i) {
            v2f av, bv;
            av.x = sr[2 * (w0 + i)] * aval;
            av.y = sr[2 * (w0 + i) + 1] * aval;
            bv.x = sB[4 * i + 2 * half][col];
            bv.y = sB[4 * i + 2 * half + 1][col];
            acc = wmma4(av, bv, acc);
        }
    }

    const float norm_sk = 0.02795084971874737f;  // 1/sqrt(128*10)
#pragma unroll
    for (int r = 0; r < 8; ++r) {
        const long n = (long)tm * 16 + half * 8 + r;
        const float val = acc[r] * norm_sk;
        const long off = (comp == 0) ? (n * 512 + col)
                                     : (n * 512 + 128 + (long)col * 3 + (comp - 1));
        out_sc[off] = val;
    }
}

// ---------------------------------------------------------------------------
// Fused edge kernel: radial MLP (WMMA) + messages + attention + atomic scatter.
// One wave (32 threads) per 16-edge tile.
// ---------------------------------------------------------------------------
__global__ __launch_bounds__(32) void edge_kernel(
    const float* __restrict__ edge_attrs, const float* __restrict__ edge_feats,
    const int* __restrict__ edge_index,   const float* __restrict__ edge_cutoff,
    const float* __restrict__ Wr1, const float* __restrict__ Wr2,
    const float* __restrict__ Wr3, const float* __restrict__ Wr4,
    const float* __restrict__ A1,  const float* __restrict__ b1,
    const float* __restrict__ ln_g, const float* __restrict__ ln_b,
    const float* __restrict__ A2,  const float* __restrict__ b2,
    const float* __restrict__ su,  const float* __restrict__ vu,
    float* __restrict__ density, float* __restrict__ M0, float* __restrict__ M1,
    int E)
{
    __shared__ float h[16][68];      // radial hidden, +4 pad (conflict-free b64)
    __shared__ float wt[16][128];    // current tpw w-group / m0a
    __shared__ float xs[16][128];    // su[sender]
    __shared__ float xv[16][384];    // vu[sender] (c*3+m)
    __shared__ float y0s[16], y1s[16][3], avals[16], cuts[16];
    __shared__ int   rcvs[16], snds[16];

    const int lane = threadIdx.x;
    const int half = lane >> 4;
    const int ll   = lane & 15;
    const long e0  = (long)blockIdx.x * 16;

    if (lane < 16) {
        const long e = e0 + lane;
        snds[lane]   = edge_index[e];
        rcvs[lane]   = edge_index[(long)E + e];
        y0s[lane]    = edge_attrs[e * 4 + 0];
        y1s[lane][0] = edge_attrs[e * 4 + 1];
        y1s[lane][1] = edge_attrs[e * 4 + 2];
        y1s[lane][2] = edge_attrs[e * 4 + 3];
        cuts[lane]   = edge_cutoff[e];
    }
    __syncthreads();

    // gather xs (32 x b128) and xv (96 x b128) per edge
    for (int ee = 0; ee < 16; ++ee) {
        const long sb = (long)snds[ee];
        const v4f* ss = (const v4f*)(su + sb * 128);
        const v4f* sv = (const v4f*)(vu + sb * 384);
        ((v4f*)&xs[ee][0])[lane] = ss[lane];
#pragma unroll
        for (int q = 0; q < 3; ++q)
            ((v4f*)&xv[ee][0])[lane + q * 32] = sv[lane + q * 32];
    }
    __syncthreads();

    // ---- radial layer 1: (16x8) @ Wr1(8x64), silu ----
    {
        const float* efb = edge_feats + (e0 + ll) * 8 + half * 2;
        float res[4][8];
#pragma unroll
        for (int tn = 0; tn < 4; ++tn) {
            const float* Wb = Wr1 + half * 2 * 64 + tn * 16 + ll;
            v8f acc = {};
#pragma unroll
            for (int k0 = 0; k0 < 8; k0 += 4) {
                v2f av, bv;
                av.x = efb[k0];
                av.y = efb[k0 + 1];
                bv.x = Wb[k0 * 64];
                bv.y = Wb[k0 * 64 + 64];
                acc = wmma4(av, bv, acc);
            }
#pragma unroll
            for (int r = 0; r < 8; ++r)
                res[tn][r] = silu_f(acc[r] * 0.3535533905932738f);  // 1/sqrt(8)
        }
#pragma unroll
        for (int tn = 0; tn < 4; ++tn)
#pragma unroll
            for (int r = 0; r < 8; ++r)
                h[half * 8 + r][tn * 16 + ll] = res[tn][r];
    }
    __syncthreads();

    // ---- radial layers 2 and 3: (16x64) @ (64x64), silu, in-place in LDS ----
    for (int layer = 0; layer < 2; ++layer) {
        const float* W  = layer ? Wr3 : Wr2;
        const float* hb = &h[ll][half * 2];
        float res[4][8];
#pragma unroll
        for (int tn = 0; tn < 4; ++tn) {
            v8f acc = tile16<64>(hb, W + half * 2 * 64 + tn * 16 + ll);
#pragma unroll
            for (int r = 0; r < 8; ++r)
                res[tn][r] = silu_f(acc[r] * 0.125f);               // 1/sqrt(64)
        }
        __syncthreads();
#pragma unroll
        for (int tn = 0; tn < 4; ++tn)
#pragma unroll
            for (int r = 0; r < 8; ++r)
                h[half * 8 + r][tn * 16 + ll] = res[tn][r];
        __syncthreads();
    }

    // ---- layer 4 streamed per 128-channel w-group + message scatter ----
    const float inv_sqrt3 = 0.5773502691896258f;
    for (int g = 0; g < 4; ++g) {
        {
            const float* hb = &h[ll][half * 2];
#pragma unroll
            for (int tn = 0; tn < 8; ++tn) {
                v8f acc = tile16<512>(hb, Wr4 + half * 2 * 512 + g * 128 + tn * 16 + ll);
#pragma unroll
                for (int r = 0; r < 8; ++r)
                    wt[half * 8 + r][tn * 16 + ll] = acc[r] * 0.125f;  // 1/sqrt(64)
            }
        }
        __syncthreads();

        if (g == 0) {
            // m0a = w0 * xs * y0   (in place in wt)
            for (int t = lane; t < 2048; t += 32) {
                const int ee = t >> 7, c = t & 127;
                wt[ee][c] *= xs[ee][c] * y0s[ee];
            }
            __syncthreads();
            // attention MLP: two half-lanes per edge, shuffle-reduce partials
            {
                const int ee = ll;
                const int cbase = half * 64;
                float z[16];
#pragma unroll
                for (int j = 0; j < 16; ++j) z[j] = (half == 0) ? b1[j] : 0.0f;
                for (int c = cbase; c < cbase + 64; ++c) {
                    const float m = wt[ee][c];
#pragma unroll
                    for (int j = 0; j < 16; ++j) z[j] += m * A1[c * 16 + j];
                }
                const long rb = (long)rcvs[ee] * 128;
                for (int c = cbase; c < cbase + 64; ++c) {
                    const float m = su[rb + c];
#pragma unroll
                    for (int j = 0; j < 16; ++j) z[j] += m * A1[(128 + c) * 16 + j];
                }
#pragma unroll
                for (int j = 0; j < 16; ++j) z[j] += __shfl_down(z[j], 16);

                if (lane < 16) {
                    float mu = 0.0f;
#pragma unroll
                    for (int j = 0; j < 16; ++j) mu += z[j];
                    mu *= (1.0f / 16.0f);
                    float var = 0.0f;
#pragma unroll
                    for (int j = 0; j < 16; ++j) { const float d = z[j] - mu; var += d * d; }
                    var *= (1.0f / 16.0f);
                    const float inv = rsqrtf(var + 1e-5f);
                    float s2 = b2[0];
#pragma unroll
                    for (int j = 0; j < 16; ++j) {
                        const float zn = (z[j] - mu) * inv * ln_g[j] + ln_b[j];
                        s2 += silu_f(zn) * A2[j];
                    }
                    const float a = __expf(s2) * cuts[ee];
                    avals[ee] = a;
                    atomicAdd(&density[rcvs[ee]], a);
                }
            }
            __syncthreads();
            for (int t = lane; t < 2048; t += 32) {
                const int ee = t >> 7, c = t & 127;
                atomicAdd(&M0[(long)rcvs[ee] * 256 + c], avals[ee] * wt[ee][c]);
            }
        } else if (g == 1) {
            // m1a = w1 * xs * y1
            for (int t = lane; t < 2048; t += 32) {
                const int ee = t >> 7, c = t & 127;
                const float v = avals[ee] * wt[ee][c] * xs[ee][c];
                const long base = (long)rcvs[ee] * 768 + (long)c * 3;
                atomicAdd(&M1[base + 0], v * y1s[ee][0]);
                atomicAdd(&M1[base + 1], v * y1s[ee][1]);
                atomicAdd(&M1[base + 2], v * y1s[ee][2]);
            }
        } else if (g == 2) {
            // m1b = w2 * xv * y0
            for (int t = lane; t < 2048; t += 32) {
                const int ee = t >> 7, c = t & 127;
                const float v = avals[ee] * wt[ee][c] * y0s[ee];
                const long base = (long)rcvs[ee] * 768 + (long)(128 + c) * 3;
                atomicAdd(&M1[base + 0], v * xv[ee][c * 3 + 0]);
                atomicAdd(&M1[base + 1], v * xv[ee][c * 3 + 1]);
                atomicAdd(&M1[base + 2], v * xv[ee][c * 3 + 2]);
            }
        } else {
            // m0b = w3 * dot(xv, y1) / sqrt(3)
            for (int t = lane; t < 2048; t += 32) {
                const int ee = t >> 7, c = t & 127;
                const float dt = xv[ee][c * 3 + 0] * y1s[ee][0]
                               + xv[ee][c * 3 + 1] * y1s[ee][1]
                               + xv[ee][c * 3 + 2] * y1s[ee][2];
                atomicAdd(&M0[(long)rcvs[ee] * 256 + 128 + c],
                          avals[ee] * wt[ee][c] * dt * inv_sqrt3);
            }
        }
        __syncthreads();
    }
}

// ---------------------------------------------------------------------------
extern "C" void kernel_launch(void* const* d_in, const int* in_sizes, int n_in,
                              void* d_out, int out_size, void* d_ws, size_t ws_size,
                              hipStream_t stream) {
    const int N = 10000, E = 160000;

    const float* node_attrs  = (const float*)d_in[0];
    const float* node_feats  = (const float*)d_in[1];
    const float* edge_attrs  = (const float*)d_in[2];
    const float* edge_feats  = (const float*)d_in[3];
    const int*   edge_index  = (const int*)  d_in[4];
    /* d_in[5] = edge_length : unused by the reference */
    const float* edge_cutoff = (const float*)d_in[6];
    const float* W_up0 = (const float*)d_in[7];
    const float* W_up1 = (const float*)d_in[8];
    const float* Wr1   = (const float*)d_in[9];
    const float* Wr2   = (const float*)d_in[10];
    const float* Wr3   = (const float*)d_in[11];
    const float* Wr4   = (const float*)d_in[12];
    const float* W_lin0 = (const float*)d_in[13];
    const float* W_lin1 = (const float*)d_in[14];
    const float* Wsk0  = (const float*)d_in[15];
    const float* Wsk1  = (const float*)d_in[16];
    const float* A1    = (const float*)d_in[17];
    const float* b1    = (const float*)d_in[18];
    const float* ln_g  = (const float*)d_in[19];
    const float* ln_b  = (const float*)d_in[20];
    const float* A2    = (const float*)d_in[21];
    const float* b2    = (const float*)d_in[22];

    float* out = (float*)d_out;              // msg: N*512 floats, then sc: N*512
    float* msg = out;
    float* sc  = out + (long)N * 512;

    float* su      = (float*)d_ws;           // N*128
    float* vu      = su + (long)N * 128;     // N*384
    float* density = vu + (long)N * 384;     // N
    float* M0      = density + N;            // N*256
    float* M1      = M0 + (long)N * 256;     // N*768

    const float inv_sqrt_C  = 0.08838834764831845f;  // 1/sqrt(128)
    const float inv_sqrt_2C = 0.0625f;               // 1/sqrt(256)

    // 1) zero atomic accumulators (density, M0, M1)
    zero_kernel<<<1024, 256, 0, stream>>>(density, (long)N * (1 + 256 + 768));

    // 2) up-projections: su = s @ W_up0 / sqrt(C); vu[n,w,m] = v[n,:,m] @ W_up1 / sqrt(C)
    gemm16_kernel<128, 1, 1, false><<<625, 256, 0, stream>>>(
        node_feats, 512, W_up0, su, 128, inv_sqrt_C, nullptr);
    for (int m = 0; m < 3; ++m)
        gemm16_kernel<128, 3, 3, false><<<625, 256, 0, stream>>>(
            node_feats + 128 + m, 512, W_up1, vu + m, 384, inv_sqrt_C, nullptr);

    // 3) skip connection -> sc part of output
    skip_kernel<<<dim3(625, 4), 256, 0, stream>>>(node_feats, node_attrs, Wsk0, Wsk1, sc);

    // 4) fused radial MLP + messages + attention + segment scatter
    edge_kernel<<<E / 16, 32, 0, stream>>>(edge_attrs, edge_feats, edge_index, edge_cutoff,
                                           Wr1, Wr2, Wr3, Wr4,
                                           A1, b1, ln_g, ln_b, A2, b2,
                                           su, vu, density, M0, M1, E);

    // 5) output linears with per-node 1/(density+1):
    //    msg[n,w,0] = out_s ; msg[n,w,1+m] = out_v[:,:,m]
    gemm16_kernel<256, 1, 4, true><<<625, 256, 0, stream>>>(
        M0, 256, W_lin0, msg, 512, inv_sqrt_2C, density);
    for (int m = 0; m < 3; ++m)
        gemm16_kernel<256, 3, 4, true><<<625, 256, 0, stream>>>(
            M1 + m, 768, W_lin1, msg + 1 + m, 512, inv_sqrt_2C, density);
}